// HeteGAT_multi_80496277061716
// MI455X (gfx1250) — compile-verified
//
#include <hip/hip_runtime.h>
#include <hip/hip_bf16.h>

typedef __attribute__((ext_vector_type(2))) float v2f;
typedef __attribute__((ext_vector_type(8))) float v8f;

#define EMB        100
#define N_NODES    2048
#define N_INPUT    256
#define N_REL      500
#define N_REL_PAD  512
#define NTILE_N    7      // ceil(100/16)

// ---------------------------------------------------------------------------
// const vector: cvec[f] = t * W1[300, f] + b1[f]
// ---------------------------------------------------------------------------
__global__ void k_constvec(const float* __restrict__ W1,
                           const float* __restrict__ b1,
                           const float* __restrict__ t,
                           float* __restrict__ cvec) {
    int f = threadIdx.x;
    if (f < EMB) cvec[f] = t[0] * W1[300 * EMB + f] + b1[f];
}

// ---------------------------------------------------------------------------
// WMMA projection: P[Mpad,100] = A[M,100] @ W[100,100] (+ cvec broadcast)
// One wave per 16x16 tile, K stepped by 4 with v_wmma_f32_16x16x4_f32.
// Branch-free inner loop: out-of-range rows/cols are CLAMPED (they produce
// duplicate values that are either never stored or multiplied by zero later).
// ---------------------------------------------------------------------------
__global__ void k_proj_wmma(const float* __restrict__ A, int M, int Mpad,
                            const float* __restrict__ W,      // [100,100] rows=K
                            const float* __restrict__ cvec,   // nullable
                            float* __restrict__ P) {
    const int lane = threadIdx.x;               // 0..31
    const int tile = blockIdx.x;
    const int m0 = (tile / NTILE_N) * 16;
    const int n0 = (tile % NTILE_N) * 16;

    const int arow  = min(m0 + (lane & 15), M - 1);   // clamp: pad rows dup last row
    const int koff  = (lane >> 4) << 1;               // 0 or 2
    const int bn_s  = n0 + (lane & 15);
    const int bn    = min(bn_s, EMB - 1);             // clamp: pad cols dup col 99

    const float* arow_p = A + (size_t)arow * EMB + koff;
    const float* wcol_p = W + bn;

    v8f c = {};
    #pragma unroll
    for (int k = 0; k < EMB; k += 4) {
        // kk = k + koff is even and row stride is even -> 8B-aligned v2f load
        v2f a = *(const v2f*)(arow_p + k);
        v2f b;
        b.x = wcol_p[(k + koff) * EMB];
        b.y = wcol_p[(k + koff + 1) * EMB];
        c = __builtin_amdgcn_wmma_f32_16x16x4_f32(false, a, false, b,
                                                  (short)0, c, false, false);
    }

    if (bn_s < EMB) {
        const float cadd = cvec ? cvec[bn_s] : 0.0f;
        const int mbase = m0 + ((lane >> 4) ? 8 : 0);
        #pragma unroll
        for (int r = 0; r < 8; ++r) {
            const int m = mbase + r;
            if (m < Mpad) P[(size_t)m * EMB + bn_s] = c[r] + cadd;
        }
    }
}

// ---------------------------------------------------------------------------
// row dot with W2: s[m] = sum_f P[m,f] * W2[f]     (one wave per row)
// ---------------------------------------------------------------------------
__global__ void k_rowdot(const float* __restrict__ P,
                         const float* __restrict__ W2,
                         float* __restrict__ s) {
    const int row = blockIdx.x;
    const int lane = threadIdx.x;
    float acc = 0.0f;
    for (int f = lane; f < EMB; f += 32) acc += P[row * EMB + f] * W2[f];
    #pragma unroll
    for (int off = 16; off > 0; off >>= 1) acc += __shfl_xor(acc, off, 32);
    if (lane == 0) s[row] = acc;
}

// ---------------------------------------------------------------------------
// per-query softmax over 2048 nodes + relation-weight histogram (LDS atomics)
//   logit = leaky_relu(sX[b] + sE[n] + sR[bias[b,n]] + b2) + mask*1e9
// ---------------------------------------------------------------------------
__global__ void k_softmax(const float* __restrict__ sX,
                          const float* __restrict__ sE,
                          const float* __restrict__ sR,
                          const int*   __restrict__ bias_mat,
                          const float* __restrict__ mask,
                          const float* __restrict__ b2,
                          float* __restrict__ Aout,
                          float* __restrict__ Wrel) {
    __shared__ float red[256];
    __shared__ float wrel[N_REL_PAD];
    const int b = blockIdx.x, tid = threadIdx.x;

    for (int i = tid; i < N_REL_PAD; i += 256) wrel[i] = 0.0f;

    const float sxb = sX[b] + b2[0];
    float lmax = -3.4e38f;
    for (int n = tid; n < N_NODES; n += 256) {
        const int r = bias_mat[b * N_NODES + n];
        float v = sxb + sE[n] + sR[r];
        v = (v > 0.0f) ? v : 0.01f * v;            // leaky_relu (slope 0.01)
        v += mask[b * N_NODES + n] * 1e9f;
        Aout[b * N_NODES + n] = v;
        lmax = fmaxf(lmax, v);
    }
    red[tid] = lmax;
    __syncthreads();
    for (int s = 128; s > 0; s >>= 1) {
        if (tid < s) red[tid] = fmaxf(red[tid], red[tid + s]);
        __syncthreads();
    }
    const float mx = red[0];
    __syncthreads();

    float lsum = 0.0f;
    for (int n = tid; n < N_NODES; n += 256) {
        const float e = __expf(Aout[b * N_NODES + n] - mx);
        Aout[b * N_NODES + n] = e;
        lsum += e;
    }
    red[tid] = lsum;
    __syncthreads();
    for (int s = 128; s > 0; s >>= 1) {
        if (tid < s) red[tid] += red[tid + s];
        __syncthreads();
    }
    const float inv = 1.0f / red[0];
    __syncthreads();

    for (int n = tid; n < N_NODES; n += 256) {
        const float a = Aout[b * N_NODES + n] * inv;
        Aout[b * N_NODES + n] = a;
        const int r = bias_mat[b * N_NODES + n];
        atomicAdd(&wrel[r], a);
    }
    __syncthreads();
    for (int i = tid; i < N_REL_PAD; i += 256)
        Wrel[b * N_REL_PAD + i] = wrel[i];
}

// ---------------------------------------------------------------------------
// final WMMA: out[256,100] = Aw[256,2048]@Ep + Wrel[256,512]@Rp + Xp + x
// Branch-free inner loops (column clamp only; rows always in range).
// ---------------------------------------------------------------------------
__global__ void k_final_wmma(const float* __restrict__ Aw,
                             const float* __restrict__ Ep,
                             const float* __restrict__ Wrel,
                             const float* __restrict__ Rp,
                             const float* __restrict__ Xp,
                             const float* __restrict__ x,
                             float* __restrict__ out) {
    const int lane = threadIdx.x;
    const int tile = blockIdx.x;
    const int m0 = (tile / NTILE_N) * 16;
    const int n0 = (tile % NTILE_N) * 16;

    const int arow  = m0 + (lane & 15);          // < 256 always
    const int koff  = (lane >> 4) << 1;
    const int bn_s  = n0 + (lane & 15);
    const int bn    = min(bn_s, EMB - 1);

    v8f c = {};
    // attention @ entity projections, K = 2048
    {
        const float* ap = Aw + (size_t)arow * N_NODES + koff;
        const float* bp = Ep + bn + (size_t)koff * EMB;
        #pragma unroll 4
        for (int k = 0; k < N_NODES; k += 4) {
            v2f a = *(const v2f*)(ap + k);
            v2f bb;
            bb.x = bp[(size_t)k * EMB];
            bb.y = bp[(size_t)(k + 1) * EMB];
            c = __builtin_amdgcn_wmma_f32_16x16x4_f32(false, a, false, bb,
                                                      (short)0, c, false, false);
        }
    }
    // relation-weight histogram @ relation projections, K = 512
    {
        const float* wp = Wrel + (size_t)arow * N_REL_PAD + koff;
        const float* bp = Rp + bn + (size_t)koff * EMB;
        #pragma unroll 4
        for (int k = 0; k < N_REL_PAD; k += 4) {
            v2f a = *(const v2f*)(wp + k);
            v2f bb;
            bb.x = bp[(size_t)k * EMB];
            bb.y = bp[(size_t)(k + 1) * EMB];
            c = __builtin_amdgcn_wmma_f32_16x16x4_f32(false, a, false, bb,
                                                      (short)0, c, false, false);
        }
    }

    if (bn_s < EMB) {
        const int mbase = m0 + ((lane >> 4) ? 8 : 0);
        #pragma unroll
        for (int r = 0; r < 8; ++r) {
            const int m = mbase + r;
            out[(size_t)m * EMB + bn_s] = c[r] + Xp[(size_t)m * EMB + bn_s]
                                               + x[(size_t)m * EMB + bn_s];
        }
    }
}

// ---------------------------------------------------------------------------
extern "C" void kernel_launch(void* const* d_in, const int* in_sizes, int n_in,
                              void* d_out, int out_size, void* d_ws, size_t ws_size,
                              hipStream_t stream) {
    const float* x    = (const float*)d_in[0];   // [256,100]
    const float* ent  = (const float*)d_in[1];   // [2048,100]
    const float* rel  = (const float*)d_in[2];   // [500,100]
    const float* t    = (const float*)d_in[3];   // [1]
    const int*   bias = (const int*)  d_in[4];   // [256,2048]
    const float* mask = (const float*)d_in[5];   // [256,2048]
    const float* W1   = (const float*)d_in[6];   // [301,100]
    const float* b1   = (const float*)d_in[7];   // [100]
    const float* W2   = (const float*)d_in[8];   // [100,1]
    const float* b2   = (const float*)d_in[9];   // [1]
    float* out = (float*)d_out;

    float* ws = (float*)d_ws;
    float* cvec = ws;                 size_t o = 128;
    float* Xp   = ws + o;             o += (size_t)N_INPUT   * EMB;
    float* Ep   = ws + o;             o += (size_t)N_NODES   * EMB;
    float* Rp   = ws + o;             o += (size_t)N_REL_PAD * EMB;
    float* sX   = ws + o;             o += N_INPUT;
    float* sE   = ws + o;             o += N_NODES;
    float* sR   = ws + o;             o += N_REL_PAD;
    float* Aw   = ws + o;             o += (size_t)N_INPUT * N_NODES;
    float* Wrel = ws + o;             o += (size_t)N_INPUT * N_REL_PAD;

    // 1) const vector  cvec = t*W1[300,:] + b1
    k_constvec<<<1, 128, 0, stream>>>(W1, b1, t, cvec);

    // 2) projections (WMMA fp32 GEMMs)
    k_proj_wmma<<<(N_INPUT   / 16) * NTILE_N, 32, 0, stream>>>(x,   N_INPUT, N_INPUT,   W1,             nullptr, Xp);
    k_proj_wmma<<<(N_NODES   / 16) * NTILE_N, 32, 0, stream>>>(ent, N_NODES, N_NODES,   W1 + 100 * EMB, cvec,    Ep);
    k_proj_wmma<<<(N_REL_PAD / 16) * NTILE_N, 32, 0, stream>>>(rel, N_REL,   N_REL_PAD, W1 + 200 * EMB, nullptr, Rp);

    // 3) scalar logit parts  s* = P @ W2
    k_rowdot<<<N_INPUT,   32, 0, stream>>>(Xp, W2, sX);
    k_rowdot<<<N_NODES,   32, 0, stream>>>(Ep, W2, sE);
    k_rowdot<<<N_REL_PAD, 32, 0, stream>>>(Rp, W2, sR);

    // 4) softmax over nodes + per-relation weight histogram
    k_softmax<<<N_INPUT, 256, 0, stream>>>(sX, sE, sR, bias, mask, b2, Aw, Wrel);

    // 5) out = Aw@Ep + Wrel@Rp + Xp + x   (WMMA)
    k_final_wmma<<<(N_INPUT / 16) * NTILE_N, 32, 0, stream>>>(Aw, Ep, Wrel, Rp, Xp, x, out);
}